// MAS_16999480557907
// MI455X (gfx1250) — compile-verified
//
#include <hip/hip_runtime.h>
#include <hip/hip_bf16.h>

#define NN 8192
#define DD 64
#define NL 3

typedef __attribute__((ext_vector_type(2))) float v2f;
typedef __attribute__((ext_vector_type(8))) float v8f;

// d_ws float layout:
//   [0 .. 389]      : per layer l (stride 130): v1[64], v2[64], c1, c2
//   [512 ..]        : t2pack[N][4]  = {t2_l0, t2_l1, t2_l2, 0} per j   (128 KB)
//   [33280 ..]      : Xsw, layout Xsw[(j>>1)*128 + col*2 + (j&1)]      (2 MB)
#define VEC_OFF 0
#define T2_OFF  512
#define XSW_OFF 33280

__device__ __forceinline__ float lrelu(float x) { return fmaxf(x, 0.2f * x); }

// ---- Kernel 1: fold attention vectors through W: v = W^T a, c = b.a ----
__global__ void mas_prep_vecs(const float* __restrict__ Ws,
                              const float* __restrict__ bWs,
                              const float* __restrict__ a1,
                              const float* __restrict__ a2,
                              float* __restrict__ ws) {
    int t = threadIdx.x;           // 192 threads
    int l = t >> 6, k = t & 63;
    float s1 = 0.f, s2 = 0.f;
    for (int d = 0; d < DD; ++d) {
        float wv = Ws[l * DD * DD + d * DD + k];
        s1 += wv * a1[l * DD + d];
        s2 += wv * a2[l * DD + d];
    }
    ws[VEC_OFF + l * 130 + k]      = s1;
    ws[VEC_OFF + l * 130 + 64 + k] = s2;
    if (k == 0) {
        float c1 = 0.f, c2 = 0.f;
        for (int d = 0; d < DD; ++d) {
            c1 += bWs[l * DD + d] * a1[l * DD + d];
            c2 += bWs[l * DD + d] * a2[l * DD + d];
        }
        ws[VEC_OFF + l * 130 + 128] = c1;
        ws[VEC_OFF + l * 130 + 129] = c2;
    }
}

// ---- Kernel 2: per-row t2 (packed float4) + swizzled X for b64 B-operand loads ----
__global__ void mas_prep_rows(const float* __restrict__ X,
                              const float* __restrict__ ba,
                              float* __restrict__ ws) {
    int i = blockIdx.x * blockDim.x + threadIdx.x;
    if (i >= NN) return;
    float xr[DD];
    const float4* xp = (const float4*)(X + (size_t)i * DD);
    #pragma unroll
    for (int q = 0; q < DD / 4; ++q) ((float4*)xr)[q] = xp[q];

    #pragma unroll
    for (int l = 0; l < NL; ++l) {
        const float* v2 = ws + VEC_OFF + l * 130 + 64;
        float d2 = ws[VEC_OFF + l * 130 + 129] + ba[l];   // c2 + ba
        for (int k = 0; k < DD; ++k) d2 += xr[k] * v2[k];
        ws[T2_OFF + i * 4 + l] = d2;
    }
    ws[T2_OFF + i * 4 + 3] = 0.f;

    float* xsw = ws + XSW_OFF + (size_t)(i >> 1) * 128 + (i & 1);
    #pragma unroll
    for (int k = 0; k < DD; ++k) xsw[k * 2] = xr[k];
}

// ---- Kernel 3: fused product-of-softmax @ X via V_WMMA_F32_16X16X4_F32 ----
// 512 blocks x 128 threads. Each block: 16 rows. 4 waves split the j range,
// combine partial C (16x64) and S (16) via ds_add_f32, then normalize.
__global__ void __launch_bounds__(128)
mas_main(const float* __restrict__ X,
         const float* __restrict__ ws,
         float* __restrict__ out) {
    __shared__ float accC[16 * 64];
    __shared__ float accS[16];
    for (int idx = threadIdx.x; idx < 16 * 64; idx += 128) accC[idx] = 0.f;
    if (threadIdx.x < 16) accS[threadIdx.x] = 0.f;
    __syncthreads();

    const int lane = threadIdx.x & 31;
    const int wv   = threadIdx.x >> 5;
    const int rowBase = blockIdx.x * 16;
    const int i = rowBase + (lane & 15);

    // t1_l[i] = X[i,:].v1_l + c1_l  (recomputed, cheaper than storing N*3)
    float t1v[NL];
    {
        float xr[DD];
        const float4* xp = (const float4*)(X + (size_t)i * DD);
        #pragma unroll
        for (int q = 0; q < DD / 4; ++q) ((float4*)xr)[q] = xp[q];
        #pragma unroll
        for (int l = 0; l < NL; ++l) {
            const float* v1 = ws + VEC_OFF + l * 130;
            float acc = ws[VEC_OFF + l * 130 + 128];       // c1
            for (int k = 0; k < DD; ++k) acc += xr[k] * v1[k];
            t1v[l] = acc;
        }
    }

    const float4* t2p = (const float4*)(ws + T2_OFF);
    const float*  xsw = ws + XSW_OFF;
    const int halfSel = (lane < 16) ? 0 : 2;   // A layout: lanes 0-15 K=0,1; 16-31 K=2,3
    const int col0    = lane & 15;

    v8f cacc0 = {}, cacc1 = {}, cacc2 = {}, cacc3 = {};
    float s_acc = 0.f;

    const int jw0 = wv * (NN / 4);             // this wave's j quarter
    for (int st = 0; st < (NN / 4) / 4; ++st) {
        int j0 = jw0 + st * 4 + halfSel;       // j0 even by construction
        float4 ta = t2p[j0];
        float4 tb = t2p[j0 + 1];
        float g0 = lrelu(t1v[0] + ta.x) + lrelu(t1v[1] + ta.y) + lrelu(t1v[2] + ta.z);
        float g1 = lrelu(t1v[0] + tb.x) + lrelu(t1v[1] + tb.y) + lrelu(t1v[2] + tb.z);
        float w0 = __expf(g0);
        float w1 = __expf(g1);
        s_acc += w0 + w1;

        v2f a; a.x = w0; a.y = w1;             // A 16x4: per lane K pair
        const v2f* brow = (const v2f*)(xsw + (size_t)(j0 >> 1) * 128);
        v2f b0 = brow[col0];                   // {X[j0][c], X[j0+1][c]}
        v2f b1 = brow[col0 + 16];
        v2f b2 = brow[col0 + 32];
        v2f b3 = brow[col0 + 48];

        cacc0 = __builtin_amdgcn_wmma_f32_16x16x4_f32(false, a, false, b0, (short)0, cacc0, false, false);
        cacc1 = __builtin_amdgcn_wmma_f32_16x16x4_f32(false, a, false, b1, (short)0, cacc1, false, false);
        cacc2 = __builtin_amdgcn_wmma_f32_16x16x4_f32(false, a, false, b2, (short)0, cacc2, false, false);
        cacc3 = __builtin_amdgcn_wmma_f32_16x16x4_f32(false, a, false, b3, (short)0, cacc3, false, false);
    }

    // cross-wave reduction in LDS (ds_add_f32)
    atomicAdd(&accS[lane & 15], s_acc);
    const int mBase = (lane < 16) ? 0 : 8;     // C/D layout: VGPR v -> M=v (+8 for hi lanes)
    #pragma unroll
    for (int v = 0; v < 8; ++v) {
        atomicAdd(&accC[(mBase + v) * 64 + col0 +  0], cacc0[v]);
        atomicAdd(&accC[(mBase + v) * 64 + col0 + 16], cacc1[v]);
        atomicAdd(&accC[(mBase + v) * 64 + col0 + 32], cacc2[v]);
        atomicAdd(&accC[(mBase + v) * 64 + col0 + 48], cacc3[v]);
    }
    __syncthreads();

    for (int idx = threadIdx.x; idx < 16 * 64; idx += 128) {
        int m = idx >> 6, n = idx & 63;
        out[(size_t)(rowBase + m) * DD + n] = accC[idx] * (1.0f / accS[m]);
    }
}

extern "C" void kernel_launch(void* const* d_in, const int* in_sizes, int n_in,
                              void* d_out, int out_size, void* d_ws, size_t ws_size,
                              hipStream_t stream) {
    const float* X   = (const float*)d_in[0];
    // d_in[1] = A : shape-only in the reference, unused
    const float* Ws  = (const float*)d_in[2];
    const float* bWs = (const float*)d_in[3];
    const float* a1  = (const float*)d_in[4];
    const float* a2  = (const float*)d_in[5];
    const float* ba  = (const float*)d_in[6];
    float* out = (float*)d_out;
    float* ws  = (float*)d_ws;

    mas_prep_vecs<<<1, 192, 0, stream>>>(Ws, bWs, a1, a2, ws);
    mas_prep_rows<<<NN / 256, 256, 0, stream>>>(X, ba, ws);
    mas_main<<<NN / 16, 128, 0, stream>>>(X, ws, out);
}